// Views_23639499997798
// MI455X (gfx1250) — compile-verified
//
#include <hip/hip_runtime.h>
#include <math.h>

#define NN 12
#define HH 224
#define WW 384
#define HWP (HH*WW)              /* 86016 */
#define PLTOT ((NN-1)*HWP)       /* 946176 */
#define RLTOT (NN*HWP)           /* 1032192 */

/* workspace layout (float offsets) */
#define ACC_OFF 0                /* [0]=pl_sum [1]=rl_sum */
#define FM_OFF  16               /* N*9 */
#define P1_OFF  128              /* 12  */
#define P2_OFF  144              /* (N-1)*12 */
#define P3_OFF  288              /* (N-1)*12 */
#define HIT_OFF 512              /* PLTOT */
#define WS_FLOATS (HIT_OFF + PLTOT)

typedef __attribute__((ext_vector_type(2))) float v2f;
typedef __attribute__((ext_vector_type(8))) float v8f;

__global__ void ws_init_kernel(float* ws) {
    int gid = blockIdx.x * 256 + threadIdx.x;
    if (gid < WS_FLOATS) ws[gid] = 0.0f;
}

/* Per-frame setup: R (Gram-Schmidt), estF, 3x3 Jacobi eigen of F^T F to
   truncate smallest singular value -> Fm; projection matrices P1/P2/P3. */
__global__ void setup_kernel(const float* __restrict__ rots,
                             const float* __restrict__ trans,
                             const float* __restrict__ focal,
                             float* __restrict__ ws) {
    int n = threadIdx.x;
    float f = fabsf(focal[0]);
    if (n == 0) {
        /* P1 = [K | 0] */
        float P1[12] = {f,0,0,0,  0,f,0,0,  0,0,1,0};
        #pragma unroll
        for (int k = 0; k < 12; ++k) ws[P1_OFF + k] = P1[k];
    }
    if (n >= NN) return;

    float v1[3], v2[3];
    #pragma unroll
    for (int k = 0; k < 3; ++k) { v1[k] = rots[n*6 + k]; v2[k] = rots[n*6 + 3 + k]; }
    float n1 = sqrtf(v1[0]*v1[0] + v1[1]*v1[1] + v1[2]*v1[2]);
    float r1[3] = { v1[0]/n1, v1[1]/n1, v1[2]/n1 };
    float dp = v2[0]*r1[0] + v2[1]*r1[1] + v2[2]*r1[2];
    float r2p[3] = { v2[0]-dp*r1[0], v2[1]-dp*r1[1], v2[2]-dp*r1[2] };
    float n2 = sqrtf(r2p[0]*r2p[0] + r2p[1]*r2p[1] + r2p[2]*r2p[2]);
    float r2[3] = { r2p[0]/n2, r2p[1]/n2, r2p[2]/n2 };
    float r3[3] = { r1[1]*r2[2]-r1[2]*r2[1], r1[2]*r2[0]-r1[0]*r2[2], r1[0]*r2[1]-r1[1]*r2[0] };
    float R[3][3];
    #pragma unroll
    for (int k = 0; k < 3; ++k) { R[k][0] = r1[k]; R[k][1] = r2[k]; R[k][2] = r3[k]; }
    float t0 = trans[n*3+0], t1 = trans[n*3+1], t2 = trans[n*3+2];

    /* KRTt = K (R^T t) */
    float w0 = r1[0]*t0 + r1[1]*t1 + r1[2]*t2;
    float w1 = r2[0]*t0 + r2[1]*t1 + r2[2]*t2;
    float w2 = r3[0]*t0 + r3[1]*t1 + r3[2]*t2;
    float k0 = f*w0, k1 = f*w1, k2 = w2;
    /* S = skew(KRTt), M2 = K*S */
    float S[3][3] = { {0.f,-k2,k1}, {k2,0.f,-k0}, {-k1,k0,0.f} };
    float M2[3][3];
    #pragma unroll
    for (int jx = 0; jx < 3; ++jx) {
        M2[0][jx] = f*S[0][jx]; M2[1][jx] = f*S[1][jx]; M2[2][jx] = S[2][jx];
    }
    /* M3 = R*M2, estF = diag(1/f,1/f,1)*M3 */
    float F[3][3];
    #pragma unroll
    for (int ix = 0; ix < 3; ++ix)
        #pragma unroll
        for (int jx = 0; jx < 3; ++jx) {
            float s = R[ix][0]*M2[0][jx] + R[ix][1]*M2[1][jx] + R[ix][2]*M2[2][jx];
            F[ix][jx] = (ix < 2) ? s/f : s;
        }
    /* A = F^T F ; Jacobi with eigenvectors */
    float A[3][3], V[3][3] = {{1,0,0},{0,1,0},{0,0,1}};
    #pragma unroll
    for (int ix = 0; ix < 3; ++ix)
        #pragma unroll
        for (int jx = 0; jx < 3; ++jx)
            A[ix][jx] = F[0][ix]*F[0][jx] + F[1][ix]*F[1][jx] + F[2][ix]*F[2][jx];
    const int JP[3] = {0,0,1}, JQ[3] = {1,2,2};
    #pragma unroll
    for (int sw = 0; sw < 8; ++sw) {
        #pragma unroll
        for (int r = 0; r < 3; ++r) {
            const int p = JP[r], q = JQ[r];
            float apq = A[p][q];
            if (fabsf(apq) > 1e-30f) {
                float th = (A[q][q]-A[p][p]) / (2.0f*apq);
                float sg = (th >= 0.0f) ? 1.0f : -1.0f;
                float tt = sg / (fabsf(th) + sqrtf(th*th + 1.0f));
                float c  = 1.0f / sqrtf(tt*tt + 1.0f);
                float s  = tt * c;
                #pragma unroll
                for (int k = 0; k < 3; ++k) { float akp=A[k][p], akq=A[k][q]; A[k][p]=c*akp-s*akq; A[k][q]=s*akp+c*akq; }
                #pragma unroll
                for (int k = 0; k < 3; ++k) { float apk=A[p][k], aqk=A[q][k]; A[p][k]=c*apk-s*aqk; A[q][k]=s*apk+c*aqk; }
                #pragma unroll
                for (int k = 0; k < 3; ++k) { float vkp=V[k][p], vkq=V[k][q]; V[k][p]=c*vkp-s*vkq; V[k][q]=s*vkp+c*vkq; }
            }
        }
    }
    int idx = 0;
    float mn = A[0][0];
    if (A[1][1] < mn) { mn = A[1][1]; idx = 1; }
    if (A[2][2] < mn) { idx = 2; }
    float v3[3];
    #pragma unroll
    for (int k = 0; k < 3; ++k)
        v3[k] = (idx == 0) ? V[k][0] : ((idx == 1) ? V[k][1] : V[k][2]);
    /* Fm = F - (F v3) v3^T */
    #pragma unroll
    for (int ix = 0; ix < 3; ++ix) {
        float Fv = F[ix][0]*v3[0] + F[ix][1]*v3[1] + F[ix][2]*v3[2];
        #pragma unroll
        for (int jx = 0; jx < 3; ++jx)
            ws[FM_OFF + n*9 + ix*3 + jx] = F[ix][jx] - Fv*v3[jx];
    }
    /* P = K [R | t] */
    float P[12] = { f*R[0][0], f*R[0][1], f*R[0][2], f*t0,
                    f*R[1][0], f*R[1][1], f*R[1][2], f*t1,
                      R[2][0],   R[2][1],   R[2][2],   t2 };
    if (n < NN-1) {
        #pragma unroll
        for (int k = 0; k < 12; ++k) ws[P2_OFF + n*12 + k] = P[k];
    }
    if (n >= 1) {
        #pragma unroll
        for (int k = 0; k < 12; ++k) ws[P3_OFF + (n-1)*12 + k] = P[k];
    }
}

/* hit[n][ii][jj] = 1 scatter (races benign: all writers store 1.0) */
__global__ void hit_kernel(const float* __restrict__ flow, float* __restrict__ ws) {
    int gid = blockIdx.x * 256 + threadIdx.x;
    if (gid >= PLTOT) return;
    int n = gid / HWP, rem = gid % HWP;
    int i = rem / WW, j = rem % WW;
    float fx = flow[(size_t)gid*2 + 0];
    float fy = flow[(size_t)gid*2 + 1];
    int jj = (int)rintf(fminf(fmaxf((float)j + fx, 0.0f), (float)(WW-1)));
    int ii = (int)rintf(fminf(fmaxf((float)i + fy, 0.0f), (float)(HH-1)));
    ws[HIT_OFF + n*HWP + ii*WW + jj] = 1.0f;
}

__global__ void rl_kernel(const float* __restrict__ flow, float* __restrict__ ws) {
    __shared__ float red[256];
    int gid = blockIdx.x * 256 + threadIdx.x;
    float val = 0.0f;
    if (gid < RLTOT) {
        int n = gid / HWP, rem = gid % HWP;
        int i = rem / WW, j = rem % WW;
        float u = (float)j - 192.0f, v = (float)i - 112.0f;
        float fx = flow[(size_t)gid*2], fy = flow[(size_t)gid*2 + 1];
        float m = (fabsf(fx) + fabsf(fy) > 20.0f) ? 1.0f : 0.0f;
        const float* F = ws + FM_OFF + n*9;
        float Fx0 = F[0]*u + F[1]*v + F[2];
        float Fx1 = F[3]*u + F[4]*v + F[5];
        float Fx2 = F[6]*u + F[7]*v + F[8];
        float ep = (u+fx)*Fx0 + (v+fy)*Fx1 + Fx2;
        val = fabsf(ep * m);
    }
    red[threadIdx.x] = val;
    __syncthreads();
    for (int s = 128; s > 0; s >>= 1) {
        if (threadIdx.x < s) red[threadIdx.x] += red[threadIdx.x + s];
        __syncthreads();
    }
    if (threadIdx.x == 0) atomicAdd(ws + 1, red[0]);
}

/* One wave per block (EXEC all-1s for WMMA). 32 pixels/wave; Gram G=C^T C of
   the per-pixel 6x4 matrix via block-diagonal V_WMMA_F32_16X16X4_F32 (4 pixels
   per tile, 2 accumulating K=4 WMMAs cover 6 rows), then per-lane 4x4 Jacobi
   for the smallest eigenvalue -> smallest singular value. */
__global__ __launch_bounds__(32) void pl_kernel(const float* __restrict__ flow,
                                                float* __restrict__ ws) {
    __shared__ float Cbuf[32*24];
    __shared__ float Dbuf[16*16];
    __shared__ float red[32];
    const int lane = threadIdx.x;
    const int bpf = HWP / 32;                 /* 2688 blocks per frame */
    const int n = blockIdx.x / bpf;
    const int rem = (blockIdx.x % bpf) * 32 + lane;
    const int i = rem / WW, j = rem % WW;

    size_t p0 = ((size_t)n * HWP + rem) * 2;
    float fx  = flow[p0],                 fy  = flow[p0 + 1];
    float fx1 = flow[p0 + (size_t)HWP*2], fy1 = flow[p0 + (size_t)HWP*2 + 1];
    float u = (float)j - 192.0f, v = (float)i - 112.0f;
    float xpx = u + fx, xpy = v + fy;
    float hitv = ws[HIT_OFF + n*HWP + rem];
    float ncx = xpx + fx1*hitv, ncy = xpy + fy1*hitv;
    float m0 = (fabsf(fx)  + fabsf(fy)  > 20.0f) ? 1.0f : 0.0f;
    float m1 = (fabsf(fx1) + fabsf(fy1) > 20.0f) ? 1.0f : 0.0f;

    const float* P1 = ws + P1_OFF;
    const float* P2 = ws + P2_OFF + n*12;
    const float* P3 = ws + P3_OFF + n*12;
    /* C rows: skew((a,b,1))[:2] @ P  ->  row0 = -P.r1 + b*P.r2 ; row1 = P.r0 - a*P.r2 */
    #pragma unroll
    for (int c = 0; c < 4; ++c) {
        Cbuf[lane*24 +  0 + c] = -P1[4+c] + v   * P1[8+c];
        Cbuf[lane*24 +  4 + c] =  P1[c]   - u   * P1[8+c];
        Cbuf[lane*24 +  8 + c] = -P2[4+c] + xpy * P2[8+c];
        Cbuf[lane*24 + 12 + c] =  P2[c]   - xpx * P2[8+c];
        Cbuf[lane*24 + 16 + c] = -P3[4+c] + ncy * P3[8+c];
        Cbuf[lane*24 + 20 + c] =  P3[c]   - ncx * P3[8+c];
    }
    __syncthreads();

    float GA[4][4];
    #pragma unroll
    for (int a = 0; a < 4; ++a)
        #pragma unroll
        for (int b = 0; b < 4; ++b) GA[a][b] = 0.0f;

    /* A-fragment layout (16x4 f32): lanes 0-15 M=lane (K=0,1 in v0,v1),
       lanes 16-31 M=lane-16 (K=2,3). A[4p+i,k] = C_p[k][i]; for G=C^T C the
       B fragment equals the A fragment under the mirrored B layout. */
    const int M  = lane & 15;
    const int pp = M >> 2, ci = M & 3;
    const int kh = lane >> 4;

    #pragma unroll
    for (int g = 0; g < 8; ++g) {
        int base = (g*4 + pp) * 24;
        v2f a0, a1;
        a0.x = Cbuf[base + (2*kh + 0)*4 + ci];
        a0.y = Cbuf[base + (2*kh + 1)*4 + ci];
        a1.x = (kh == 0) ? Cbuf[base + 16 + ci] : 0.0f;   /* C rows 4,5 */
        a1.y = (kh == 0) ? Cbuf[base + 20 + ci] : 0.0f;
        v8f d = {0.f,0.f,0.f,0.f,0.f,0.f,0.f,0.f};
        d = __builtin_amdgcn_wmma_f32_16x16x4_f32(false, a0, false, a0,
                                                  (short)0, d, false, false);
        d = __builtin_amdgcn_wmma_f32_16x16x4_f32(false, a1, false, a1,
                                                  (short)0, d, false, false);
        /* D layout: vgpr r, lanes 0-15 -> (M=r, N=lane); lanes 16-31 -> (M=r+8, N=lane-16) */
        int mbase = (lane < 16) ? 0 : 8;
        int ncol  = lane & 15;
        #pragma unroll
        for (int r = 0; r < 8; ++r) Dbuf[(mbase + r)*16 + ncol] = d[r];
        __syncthreads();
        if ((lane >> 2) == g) {            /* lanes 4g..4g+3 own these 4 pixels */
            int p = lane & 3;
            #pragma unroll
            for (int a = 0; a < 4; ++a)
                #pragma unroll
                for (int b = 0; b < 4; ++b)
                    GA[a][b] = Dbuf[(4*p + a)*16 + (4*p + b)];
        }
        __syncthreads();
    }

    /* smallest eigenvalue of 4x4 SPD Gram via unrolled cyclic Jacobi */
    const int JP[6] = {0,0,0,1,1,2}, JQ[6] = {1,2,3,2,3,3};
    #pragma unroll
    for (int sw = 0; sw < 5; ++sw) {
        #pragma unroll
        for (int t = 0; t < 6; ++t) {
            const int p = JP[t], q = JQ[t];
            float apq = GA[p][q];
            if (fabsf(apq) > 1e-28f) {
                float th = (GA[q][q] - GA[p][p]) / (2.0f*apq);
                float sg = (th >= 0.0f) ? 1.0f : -1.0f;
                float tt = sg / (fabsf(th) + sqrtf(th*th + 1.0f));
                float c  = 1.0f / sqrtf(tt*tt + 1.0f);
                float s  = tt * c;
                #pragma unroll
                for (int k = 0; k < 4; ++k) { float akp=GA[k][p], akq=GA[k][q]; GA[k][p]=c*akp-s*akq; GA[k][q]=s*akp+c*akq; }
                #pragma unroll
                for (int k = 0; k < 4; ++k) { float apk=GA[p][k], aqk=GA[q][k]; GA[p][k]=c*apk-s*aqk; GA[q][k]=s*apk+c*aqk; }
            }
        }
    }
    float lmin = fminf(fminf(GA[0][0], GA[1][1]), fminf(GA[2][2], GA[3][3]));
    float sv = sqrtf(fmaxf(lmin, 0.0f));

    red[lane] = sv * m0 * m1;
    __syncthreads();
    if (lane == 0) {
        float s = 0.0f;
        #pragma unroll
        for (int t = 0; t < 32; ++t) s += red[t];
        atomicAdd(ws + 0, s);
    }
}

__global__ void finalize_kernel(const float* __restrict__ ws, float* __restrict__ out) {
    out[0] = ws[0] * (1.0f / (float)PLTOT) + 0.001f * ws[1] * (1.0f / (float)RLTOT);
}

extern "C" void kernel_launch(void* const* d_in, const int* in_sizes, int n_in,
                              void* d_out, int out_size, void* d_ws, size_t ws_size,
                              hipStream_t stream) {
    const float* flow  = (const float*)d_in[0];
    const float* rots  = (const float*)d_in[1];
    const float* trans = (const float*)d_in[2];
    const float* focal = (const float*)d_in[3];
    float* ws  = (float*)d_ws;
    float* out = (float*)d_out;

    ws_init_kernel<<<(WS_FLOATS + 255)/256, 256, 0, stream>>>(ws);
    setup_kernel<<<1, 32, 0, stream>>>(rots, trans, focal, ws);
    hit_kernel<<<(PLTOT + 255)/256, 256, 0, stream>>>(flow, ws);
    rl_kernel<<<(RLTOT + 255)/256, 256, 0, stream>>>(flow, ws);
    pl_kernel<<<(NN-1)*(HWP/32), 32, 0, stream>>>(flow, ws);
    finalize_kernel<<<1, 1, 0, stream>>>(ws, out);
}